// Corr_11553462026420
// MI455X (gfx1250) — compile-verified
//
#include <hip/hip_runtime.h>
#include <hip/hip_bf16.h>

typedef float v2f __attribute__((ext_vector_type(2)));
typedef float v8f __attribute__((ext_vector_type(8)));

#define NCLS 4
#define CIN  32
#define HW   6400   // 80*80
#define NT   400    // HW/16
#define NB   2

__device__ __forceinline__ v2f load2(const float* p) {
  return *reinterpret_cast<const v2f*>(p);
}

// ---------------------------------------------------------------------------
// prep: f1 = 0.5*(w1*feat + b1)  (softmax scale folded in), f2 = w2*feat + b2,
//       V  = bilinear-upsampled(out, align_corners) ; all stored [n][p][4]
// ---------------------------------------------------------------------------
__global__ void prep_kernel(const float* __restrict__ feat,
                            const float* __restrict__ out40,
                            const float* __restrict__ w1, const float* __restrict__ b1,
                            const float* __restrict__ w2, const float* __restrict__ b2,
                            float* __restrict__ f1buf, float* __restrict__ f2buf,
                            float* __restrict__ vbuf) {
  int idx = blockIdx.x * blockDim.x + threadIdx.x;
  if (idx >= NB * HW) return;
  int n = idx / HW, p = idx - n * HW;

  float a1[NCLS], a2[NCLS];
#pragma unroll
  for (int o = 0; o < NCLS; ++o) { a1[o] = b1[o]; a2[o] = b2[o]; }
  const float* fp = feat + n * CIN * HW + p;
#pragma unroll 4
  for (int c = 0; c < CIN; ++c) {
    float fv = fp[c * HW];
#pragma unroll
    for (int o = 0; o < NCLS; ++o) {
      a1[o] = fmaf(w1[o * CIN + c], fv, a1[o]);
      a2[o] = fmaf(w2[o * CIN + c], fv, a2[o]);
    }
  }
#pragma unroll
  for (int o = 0; o < NCLS; ++o) {
    f1buf[idx * 4 + o] = 0.5f * a1[o];   // fold 1/sqrt(nclass)=0.5 into f1
    f2buf[idx * 4 + o] = a2[o];
  }

  // align_corners bilinear: 40x40 -> 80x80, coord = dst * (39/79)
  int y = p / 80, x = p - y * 80;
  float fy = (float)y * (39.0f / 79.0f);
  float fx = (float)x * (39.0f / 79.0f);
  int y0 = min((int)fy, 39), x0 = min((int)fx, 39);
  int y1 = min(y0 + 1, 39),  x1 = min(x0 + 1, 39);
  float wy = fy - (float)y0, wx = fx - (float)x0;
  float c00 = (1.0f - wy) * (1.0f - wx), c01 = (1.0f - wy) * wx;
  float c10 = wy * (1.0f - wx),          c11 = wy * wx;
#pragma unroll
  for (int o = 0; o < NCLS; ++o) {
    const float* src = out40 + (n * NCLS + o) * 1600;
    float v = src[y0 * 40 + x0] * c00 + src[y0 * 40 + x1] * c01 +
              src[y1 * 40 + x0] * c10 + src[y1 * 40 + x1] * c11;
    vbuf[idx * 4 + o] = v;
  }
}

// ---------------------------------------------------------------------------
// rowstats: per 16-row p-tile, sweep all q-tiles with WMMA f32 16x16x4.
// Pass 1: row max (half-wave butterfly). Pass 2: Z = sum exp(S - m).
// Store (m, 1/Z) per p.
// ---------------------------------------------------------------------------
__global__ __launch_bounds__(128)
void rowstats_kernel(const float* __restrict__ f1buf,
                     const float* __restrict__ f2buf,
                     float2* __restrict__ mzbuf) {
  int lane = threadIdx.x & 31;
  int tile = blockIdx.x * 4 + (threadIdx.x >> 5);   // 0..799
  int n = tile / NT, pt = tile - n * NT;
  int l15 = lane & 15, half = lane >> 4;

  const float* f1 = f1buf + n * HW * 4;
  const float* f2 = f2buf + n * HW * 4;
  int p0 = pt * 16;

  // A operand: lanes 0-15 hold (M=lane, K=0..1); lanes 16-31 (M=lane-16, K=2..3)
  v2f a = load2(f1 + (p0 + l15) * 4 + half * 2);

  float mrow[8];
#pragma unroll
  for (int r = 0; r < 8; ++r) mrow[r] = -3.4e38f;

  for (int j = 0; j < NT; ++j) {
    v2f b = load2(f2 + (j * 16 + l15) * 4 + half * 2);
    v8f cz = {};
    v8f c = __builtin_amdgcn_wmma_f32_16x16x4_f32(false, a, false, b,
                                                  (short)0, cz, false, false);
#pragma unroll
    for (int r = 0; r < 8; ++r) mrow[r] = fmaxf(mrow[r], c[r]);
  }
#pragma unroll
  for (int r = 0; r < 8; ++r) {
#pragma unroll
    for (int mask = 1; mask <= 8; mask <<= 1)
      mrow[r] = fmaxf(mrow[r], __shfl_xor(mrow[r], mask, 32));
  }

  float srow[8];
#pragma unroll
  for (int r = 0; r < 8; ++r) srow[r] = 0.0f;

  for (int j = 0; j < NT; ++j) {
    v2f b = load2(f2 + (j * 16 + l15) * 4 + half * 2);
    v8f cz = {};
    v8f c = __builtin_amdgcn_wmma_f32_16x16x4_f32(false, a, false, b,
                                                  (short)0, cz, false, false);
#pragma unroll
    for (int r = 0; r < 8; ++r) srow[r] += __expf(c[r] - mrow[r]);
  }
#pragma unroll
  for (int r = 0; r < 8; ++r) {
#pragma unroll
    for (int mask = 1; mask <= 8; mask <<= 1)
      srow[r] += __shfl_xor(srow[r], mask, 32);
  }

  if (l15 == 0) {
    int pbase = p0 + half * 8;   // D layout: half 0 -> M=r, half 1 -> M=8+r
#pragma unroll
    for (int r = 0; r < 8; ++r)
      mzbuf[n * HW + pbase + r] = make_float2(mrow[r], 1.0f / srow[r]);
  }
}

// ---------------------------------------------------------------------------
// attn: per 16-col q-tile, sweep p-tiles: S tile via WMMA, w = exp(S-m)/Z,
// acc[c] += V[c,p]*w over 8 rows/lane; halves combined via shfl_xor(16).
// ---------------------------------------------------------------------------
__global__ __launch_bounds__(128)
void attn_kernel(const float* __restrict__ f1buf,
                 const float* __restrict__ f2buf,
                 const float* __restrict__ vbuf,
                 const float2* __restrict__ mzbuf,
                 float* __restrict__ outp) {
  int lane = threadIdx.x & 31;
  int tile = blockIdx.x * 4 + (threadIdx.x >> 5);   // 0..799
  int n = tile / NT, qt = tile - n * NT;
  int l15 = lane & 15, half = lane >> 4;

  const float*  f1 = f1buf + n * HW * 4;
  const float*  f2 = f2buf + n * HW * 4;
  const float4* V4 = reinterpret_cast<const float4*>(vbuf) + n * HW;
  const float2* mz = mzbuf + n * HW;

  int q0 = qt * 16;
  v2f b = load2(f2 + (q0 + l15) * 4 + half * 2);

  float acc[NCLS] = {0.0f, 0.0f, 0.0f, 0.0f};

  for (int i = 0; i < NT; ++i) {
    int p0 = i * 16;
    v2f a = load2(f1 + (p0 + l15) * 4 + half * 2);
    v8f cz = {};
    v8f c = __builtin_amdgcn_wmma_f32_16x16x4_f32(false, a, false, b,
                                                  (short)0, cz, false, false);
    int pb = p0 + half * 8;
#pragma unroll
    for (int r = 0; r < 8; ++r) {
      float2 m  = mz[pb + r];
      float  w  = __expf(c[r] - m.x) * m.y;
      float4 vv = V4[pb + r];
      acc[0] = fmaf(vv.x, w, acc[0]);
      acc[1] = fmaf(vv.y, w, acc[1]);
      acc[2] = fmaf(vv.z, w, acc[2]);
      acc[3] = fmaf(vv.w, w, acc[3]);
    }
  }
#pragma unroll
  for (int ch = 0; ch < NCLS; ++ch)
    acc[ch] += __shfl_xor(acc[ch], 16, 32);

  if (lane < 16) {
    int q = q0 + lane;
#pragma unroll
    for (int ch = 0; ch < NCLS; ++ch)
      outp[(n * NCLS + ch) * HW + q] = acc[ch];
  }
}

// ---------------------------------------------------------------------------
extern "C" void kernel_launch(void* const* d_in, const int* in_sizes, int n_in,
                              void* d_out, int out_size, void* d_ws, size_t ws_size,
                              hipStream_t stream) {
  const float* feat  = (const float*)d_in[0];  // [2,32,80,80]
  const float* out40 = (const float*)d_in[1];  // [2,4,40,40]
  const float* w1    = (const float*)d_in[2];  // [4,32]
  const float* b1    = (const float*)d_in[3];  // [4]
  const float* w2    = (const float*)d_in[4];  // [4,32]
  const float* b2    = (const float*)d_in[5];  // [4]

  float* ws     = (float*)d_ws;
  float* f1buf  = ws;                    // 2*6400*4 = 51200 floats
  float* f2buf  = ws + 51200;            // 51200 floats
  float* vbuf   = ws + 102400;           // 51200 floats
  float2* mzbuf = (float2*)(ws + 153600);// 2*6400 float2 = 25600 floats

  prep_kernel<<<(NB * HW + 255) / 256, 256, 0, stream>>>(
      feat, out40, w1, b1, w2, b2, f1buf, f2buf, vbuf);

  // 800 tiles, 4 waves (128 threads) per block
  rowstats_kernel<<<200, 128, 0, stream>>>(f1buf, f2buf, mzbuf);

  attn_kernel<<<200, 128, 0, stream>>>(f1buf, f2buf, vbuf, mzbuf, (float*)d_out);
}